// ranking_loss_7155415515603
// MI455X (gfx1250) — compile-verified
//
#include <hip/hip_runtime.h>

typedef __attribute__((ext_vector_type(2))) float v2f;
typedef __attribute__((ext_vector_type(8))) float v8f;

#define K_DIM   2400
#define MARGIN  1.0f
#define EPS_F   1e-6f
#define ROWS_PER_BLOCK 128   // 8 waves * 16 rows
#define KC      32           // K columns per pipeline stage (128 B/row)
#define STAGES  (K_DIM / KC) // 75
#define LPITCH  36           // padded LDS row pitch (floats): conflict-free b64 reads

// workspace layout (in floats)
#define WS_BVEC 0            // 2400 floats: anchor + eps
#define WS_BSQ  2400         // 1 float:   ||b||^2
#define WS_DAP  2401         // up to 15:  d_ap[j]
#define WS_PART 2416         // per-block partial sums

// ---------------------------------------------------------------------------
// Kernel 1: b = anchor + eps, ||b||^2, and d_ap[j] = ||anchor - pos_j + eps||
// ---------------------------------------------------------------------------
__global__ void prep_kernel(const float* __restrict__ anchor,
                            const float* __restrict__ pos,
                            float* __restrict__ ws, int P) {
    __shared__ float red[256];
    const int tid = threadIdx.x;

    float s = 0.f;
    for (int i = tid; i < K_DIM; i += 256) {
        float b = anchor[i] + EPS_F;
        ws[WS_BVEC + i] = b;
        s += b * b;
    }
    red[tid] = s; __syncthreads();
    for (int o = 128; o > 0; o >>= 1) {
        if (tid < o) red[tid] += red[tid + o];
        __syncthreads();
    }
    if (tid == 0) ws[WS_BSQ] = red[0];
    __syncthreads();

    for (int j = 0; j < P; ++j) {
        float t = 0.f;
        for (int i = tid; i < K_DIM; i += 256) {
            float d = anchor[i] - pos[j * K_DIM + i] + EPS_F;
            t += d * d;
        }
        red[tid] = t; __syncthreads();
        for (int o = 128; o > 0; o >>= 1) {
            if (tid < o) red[tid] += red[tid + o];
            __syncthreads();
        }
        if (tid == 0) ws[WS_DAP + j] = sqrtf(red[0]);
        __syncthreads();
    }
}

// ---------------------------------------------------------------------------
// Kernel 2: async-copy negatives tiles into LDS (double buffered, per-wave),
// b-vector resident in LDS; WMMA computes b . n_i for 16 rows/wave, VALU FMAs
// accumulate ||n_i||^2 on the same registers. Steady-state global traffic is
// only the ASYNCcnt-tracked negatives stream.
// ---------------------------------------------------------------------------
__global__ void main_kernel(const float* __restrict__ neg,
                            const float* __restrict__ ws,
                            float* __restrict__ partials,
                            int N, int numFull, int P) {
    __shared__ __align__(16) float tile[8][2][16 * LPITCH];
    __shared__ __align__(16) float bsh[K_DIM];
    __shared__ float red[256];

    const int tid  = threadIdx.x;
    const int lane = tid & 31;
    const int wv   = tid >> 5;
    const int h    = lane >> 4;     // lane half (selects K pair within chunk of 4)
    const int m    = lane & 15;     // row within 16-row tile

    const int wave_row0 = blockIdx.x * ROWS_PER_BLOCK + wv * 16;
    const int grow = wave_row0 + m;

    // LDS byte offsets of this wave's two buffers (flat addr low 32 bits)
    const unsigned tb[2] = { (unsigned)(size_t)&tile[wv][0][0],
                             (unsigned)(size_t)&tile[wv][1][0] };

    // async-load lane mapping: 4 rows x 128 contiguous bytes per instruction
    const int rsel  = lane >> 3;        // row within group of 4
    const int csel4 = (lane & 7) * 4;   // float offset of this lane's 16B granule

    // Issue one stage (16 rows x KC floats) as 4 async b128 instructions.
    auto issue_stage = [&](int k0, unsigned tbase) {
#pragma unroll
        for (int j = 0; j < 4; ++j) {
            int rr = wave_row0 + 4 * j + rsel;
            int cr = (rr < N) ? rr : (N - 1);      // clamp tail rows (safe reads)
            unsigned long long g =
                (unsigned long long)(size_t)(neg + (size_t)cr * K_DIM + k0 + csel4);
            unsigned l = tbase + (unsigned)((4 * j + rsel) * LPITCH + csel4) * 4u;
            asm volatile("global_load_async_to_lds_b128 %0, %1, off"
                         :: "v"(l), "v"(g) : "memory");
        }
    };

    // Kick off stage 0 while the block stages the b-vector into LDS.
    issue_stage(0, tb[0]);
    for (int i = tid; i < K_DIM; i += 256) bsh[i] = ws[WS_BVEC + i];
    __syncthreads();

    v8f c0 = {0.f,0.f,0.f,0.f,0.f,0.f,0.f,0.f};
    v8f c1 = {0.f,0.f,0.f,0.f,0.f,0.f,0.f,0.f};
    float sq = 0.f;
    const float* bvp = bsh + 2 * h;

    for (int s = 0; s < STAGES; ++s) {
        if (s + 1 < STAGES) {
            // previous-buffer ds reads must be in VGPRs before async overwrite
            asm volatile("s_wait_dscnt 0x0" ::: "memory");
            issue_stage((s + 1) * KC, tb[(s + 1) & 1]);
            asm volatile("s_wait_asynccnt 0x4" ::: "memory");  // stage s landed
        } else {
            asm volatile("s_wait_asynccnt 0x0" ::: "memory");  // last stage landed
        }

        const float* lrow = &tile[wv][s & 1][m * LPITCH + 2 * h];
        const float* bk   = bvp + s * KC;
#pragma unroll
        for (int k = 0; k < KC; k += 8) {
            float2 a = *(const float2*)(lrow + k);
            float2 b = *(const float2*)(bk + k);
            v2f av = {a.x, a.y};
            v2f bw = {b.x, b.y};
            c0 = __builtin_amdgcn_wmma_f32_16x16x4_f32(false, av, false, bw,
                                                       (short)0, c0, false, false);
            sq = fmaf(a.x, a.x, sq);
            sq = fmaf(a.y, a.y, sq);

            float2 a2 = *(const float2*)(lrow + k + 4);
            float2 b2 = *(const float2*)(bk + k + 4);
            v2f av2 = {a2.x, a2.y};
            v2f bw2 = {b2.x, b2.y};
            c1 = __builtin_amdgcn_wmma_f32_16x16x4_f32(false, av2, false, bw2,
                                                       (short)0, c1, false, false);
            sq = fmaf(a2.x, a2.x, sq);
            sq = fmaf(a2.y, a2.y, sq);
        }
    }
    v8f c = c0 + c1;

    // Full ||n||^2 for row m: this lane covers K%4 in {2h, 2h+1}; partner lane
    // (xor 16) covers the other pair.
    float sq_full = sq + __shfl_xor(sq, 16, 32);

    // Dot for row m: lanes 0-15 accumulators hold rows 0-7, lanes 16-31 hold 8-15
    // (every lane in a half holds all 8 since B columns are uniform).
    const int mi = m & 7;
    float own = c[0];
    own = (mi == 1) ? c[1] : own;
    own = (mi == 2) ? c[2] : own;
    own = (mi == 3) ? c[3] : own;
    own = (mi == 4) ? c[4] : own;
    own = (mi == 5) ? c[5] : own;
    own = (mi == 6) ? c[6] : own;
    own = (mi == 7) ? c[7] : own;
    float up  = __shfl(own, (lane & 15) + 16, 32);  // row m dot from upper half
    float dot = (m < 8) ? own : up;

    float contrib = 0.f;
    if (lane < 16 && grow < numFull) {
        float d2 = sq_full + ws[WS_BSQ] - 2.f * dot;
        d2 = fmaxf(d2, 0.f);
        float dan = sqrtf(d2);
        float dap = ws[WS_DAP + (grow % P)];
        contrib = fmaxf(0.f, dap - dan + MARGIN);
    }

    red[tid] = contrib; __syncthreads();
    for (int o = 128; o > 0; o >>= 1) {
        if (tid < o) red[tid] += red[tid + o];
        __syncthreads();
    }
    if (tid == 0) partials[blockIdx.x] = red[0];
}

// ---------------------------------------------------------------------------
// Kernel 3: deterministic final reduction of per-block partials.
// ---------------------------------------------------------------------------
__global__ void final_kernel(const float* __restrict__ partials,
                             float* __restrict__ out, int nblocks) {
    __shared__ float red[256];
    const int tid = threadIdx.x;
    float s = 0.f;
    for (int i = tid; i < nblocks; i += 256) s += partials[i];
    red[tid] = s; __syncthreads();
    for (int o = 128; o > 0; o >>= 1) {
        if (tid < o) red[tid] += red[tid + o];
        __syncthreads();
    }
    if (tid == 0) out[0] = red[0];
}

// ---------------------------------------------------------------------------
extern "C" void kernel_launch(void* const* d_in, const int* in_sizes, int n_in,
                              void* d_out, int out_size, void* d_ws, size_t ws_size,
                              hipStream_t stream) {
    const float* anchor = (const float*)d_in[0];
    const float* pos    = (const float*)d_in[1];
    const float* neg    = (const float*)d_in[2];
    float* ws  = (float*)d_ws;
    float* out = (float*)d_out;

    const int P = in_sizes[1] / K_DIM;              // 5
    const int N = in_sizes[2] / K_DIM;              // 100000
    const int numFull = (N / P) * P;
    const int nblocks = (N + ROWS_PER_BLOCK - 1) / ROWS_PER_BLOCK;

    prep_kernel<<<1, 256, 0, stream>>>(anchor, pos, ws, P);
    main_kernel<<<nblocks, 256, 0, stream>>>(neg, ws, ws + WS_PART, N, numFull, P);
    final_kernel<<<1, 256, 0, stream>>>(ws + WS_PART, out, nblocks);
}